// SamplingAttention_RA_25280177504719
// MI455X (gfx1250) — compile-verified
//
#include <hip/hip_runtime.h>
#include <hip/hip_bf16.h>
#include <math.h>

// ---- problem constants ----
#define DMODEL 256
#define MHEAD  8
#define PPTS   4
#define FLVL   4
#define LQ     300
#define NB     8
#define STOT   13294   // 100*100 + 50*50 + 25*25 + 13*13
#define NM     (NB*MHEAD)   // 64
#define CH     (DMODEL/MHEAD) // 32

typedef float v2f __attribute__((ext_vector_type(2)));
typedef float v8f __attribute__((ext_vector_type(8)));

// =====================================================================
// Kernel A: value conv  value_t[n][s][o] = sum_c W[o][c]*k[n][c][s] + b[o]
// WMMA f32 16x16x4:  M axis = s (so stores are lane-contiguous in o),
//                    N axis = o, K axis = c.
// Workgroup = 4 waves, tile = 64(s) x 64(o); K staged in 64-chunks in LDS.
// =====================================================================
__global__ __launch_bounds__(128) void value_conv_kernel(
    const float* __restrict__ k, const float* __restrict__ Wc,
    const float* __restrict__ bc, const unsigned char* __restrict__ mask,
    float* __restrict__ value_t)
{
    const int s_tile = blockIdx.x;   // 208 tiles over S
    const int o_tile = blockIdx.y;   // 4 tiles over D
    const int n      = blockIdx.z;   // batch
    const int tid  = threadIdx.x;
    const int lane = tid & 31;
    const int wave = tid >> 5;
    const int s_base = s_tile * 64;
    const int o_base = o_tile * 64;

    // pad rows to 68 floats: stride%64==4 -> conflict-free fragment reads,
    // and 16B-aligned float4 stores into Wlds.
    __shared__ float Klds[64 * 68];   // [s_local][c_local]
    __shared__ float Wlds[64 * 68];   // [o_local][c_local]

    const int row = lane & 15;         // M-row (A) / N-col (B) within fragment
    const int kk  = (lane >> 4) << 1;  // K sub-offset: lanes 0-15 -> K0,K1 ; 16-31 -> K2,K3

    v8f acc[4] = {v8f{}, v8f{}, v8f{}, v8f{}};

    const float* kn = k + (size_t)n * DMODEL * STOT;

    for (int c0 = 0; c0 < DMODEL; c0 += 64) {
        __syncthreads();   // protect LDS reuse across chunks
        // ---- stage W panel: Wlds[o][c] = Wc[o_base+o][c0+c]  (row-major copy)
        for (int idx = tid; idx < 64 * 16; idx += 128) {
            const int o_l = idx >> 4, t = idx & 15;
            const float4 w4 = *(const float4*)&Wc[(size_t)(o_base + o_l) * DMODEL + c0 + t * 4];
            *(float4*)&Wlds[o_l * 68 + t * 4] = w4;
        }
        // ---- stage K panel transposed: Klds[s][c] = k[n][c0+c][s_base+s]
        // single-use stream -> non-temporal loads keep L2 for value_t
        for (int idx = tid; idx < 64 * 16; idx += 128) {
            const int c_l = idx >> 4, s4 = idx & 15;
            const float* src = kn + (size_t)(c0 + c_l) * STOT + s_base + s4 * 4;
#pragma unroll
            for (int i = 0; i < 4; ++i) {
                const int s = s_base + s4 * 4 + i;
                const float v = (s < STOT) ? __builtin_nontemporal_load(&src[i]) : 0.0f;
                Klds[(s4 * 4 + i) * 68 + c_l] = v;
            }
        }
        __syncthreads();

        const float* Abase = &Klds[(wave * 16 + row) * 68 + kk];
#pragma unroll
        for (int cc = 0; cc < 64; cc += 4) {
            const v2f a = *(const v2f*)(Abase + cc);   // A: 16x4 slice of k^T (s x c)
#pragma unroll
            for (int t = 0; t < 4; ++t) {              // B: 4x16 slice of W^T (c x o)
                const v2f b = *(const v2f*)&Wlds[(t * 16 + row) * 68 + cc + kk];
                acc[t] = __builtin_amdgcn_wmma_f32_16x16x4_f32(
                    false, a, false, b, (short)0, acc[t], false, false);
            }
        }
    }

    // ---- store: C layout (16x16 f32): VGPR v, lanes 0-15 -> M=v, lanes 16-31 -> M=v+8
    const int hi = (lane >> 4) * 8;
#pragma unroll
    for (int t = 0; t < 4; ++t) {
        const int o = o_base + t * 16 + row;
        const float bo = bc[o];
#pragma unroll
        for (int v = 0; v < 8; ++v) {
            const int s = s_base + wave * 16 + v + hi;
            if (s < STOT) {
                float val = acc[t][v] + bo;
                if (mask[(size_t)n * STOT + s]) val = 0.0f;
                value_t[((size_t)n * STOT + s) * DMODEL + o] = val;  // channels-last
            }
        }
    }
}

// =====================================================================
// Kernel B: per (l, n, m): 48 GEMV dots (16 weight logits + 32 offsets),
// softmax, grid transform, write attn tensor + sampler params.
// Block = 64 threads (2 waves) per (l, nm).
// =====================================================================
__device__ __forceinline__ float sigm(float x) { return 1.0f / (1.0f + __expf(-x)); }

__global__ __launch_bounds__(64) void param_kernel(
    const float* __restrict__ q, const float* __restrict__ pos_centers,
    const float* __restrict__ valid_scales,
    const float* __restrict__ loc_w, const float* __restrict__ loc_b,
    const float* __restrict__ wgt_w, const float* __restrict__ wgt_b,
    float* __restrict__ attn, float* __restrict__ samp)
{
    const int bid = blockIdx.x;        // l*64 + nm
    const int l = bid >> 6, nm = bid & 63;
    const int n = nm >> 3,  m  = nm & 7;
    const int j = threadIdx.x;

    __shared__ float raw[48];          // [0..15] wgt logits, [16..47] raw grids

    const float* qrow = q + ((size_t)l * NB + n) * DMODEL;
    if (j < 48) {
        const float* wrow;
        float s;
        if (j < 16) { wrow = wgt_w + (size_t)(m * 16 + j) * DMODEL;        s = wgt_b[m * 16 + j]; }
        else        { const int jj = j - 16;
                      wrow = loc_w + (size_t)(m * 32 + jj) * DMODEL;       s = loc_b[m * 32 + jj]; }
        for (int d = 0; d < DMODEL; ++d) s = fmaf(qrow[d], wrow[d], s);
        raw[j] = s;
    }
    __syncthreads();

    if (j < 16) {
        // softmax over the 16 logits
        float mx = raw[0];
#pragma unroll
        for (int t = 1; t < 16; ++t) mx = fmaxf(mx, raw[t]);
        float sum = 0.0f;
#pragma unroll
        for (int t = 0; t < 16; ++t) sum += __expf(raw[t] - mx);
        const float w = __expf(raw[j] - mx) / sum;

        const int f = j >> 2, p = j & 3;            // j = f*P + p
        const float rx = raw[16 + p * 8 + f * 2 + 0]; // loc idx = p*FL*2 + f*2 + xy
        const float ry = raw[16 + p * 8 + f * 2 + 1];

        const float* pc = pos_centers + ((size_t)l * NB + n) * 4;
        const float wh_x = sigm(pc[2]), wh_y = sigm(pc[3]);
        const float bx = sigm(pc[0]) - 0.5f * wh_x;
        const float by = sigm(pc[1]) - 0.5f * wh_y;
        const float gpx = (p + 0.5f) * (1.0f / PPTS) * wh_x;
        const float gpy = (m + 0.5f) * (1.0f / MHEAD) * wh_y;
        const float vsx = 2.0f * valid_scales[((size_t)n * FLVL + f) * 2 + 0];
        const float vsy = 2.0f * valid_scales[((size_t)n * FLVL + f) * 2 + 1];
        const float gx = (rx * wh_x * (1.0f / PPTS) + bx + gpx) * vsx - 1.0f;
        const float gy = (ry * wh_y * (1.0f / PPTS) + by + gpy) * vsy - 1.0f;

        // attn: [N, L, FL, M, P, 3]
        const size_t ai = (((((size_t)n * LQ + l) * FLVL + f) * MHEAD + m) * PPTS + p) * 3;
        attn[ai + 0] = (gx + 1.0f) * 0.5f;
        attn[ai + 1] = (gy + 1.0f) * 0.5f;
        attn[ai + 2] = w;
        // sampler params: [l*64+nm][16 pts][gx,gy,w]
        const size_t si = ((size_t)bid * 16 + j) * 3;
        samp[si + 0] = gx; samp[si + 1] = gy; samp[si + 2] = w;
    }
}

// =====================================================================
// Kernel C: bilinear sampling + weighted sum.
// One wave per (l, nm); lane = channel (CH == 32 == wave32).
// value_t is channels-last, so each tap reads 128 contiguous bytes/wave,
// and the whole 109 MB value tensor sits in the 192 MB L2.
// =====================================================================
__global__ __launch_bounds__(256) void sample_kernel(
    const float* __restrict__ value_t, const float* __restrict__ samp,
    float* __restrict__ pre)
{
    const int wid  = blockIdx.x * 8 + (threadIdx.x >> 5);   // l*64 + nm
    const int lane = threadIdx.x & 31;
    const int l = wid >> 6, nm = wid & 63;
    const int n = nm >> 3,  m  = nm & 7;

    const int OFF[4] = {0, 10000, 12500, 13125};
    const int HH [4] = {100, 50, 25, 13};

    const float* base = value_t + ((size_t)n * STOT) * DMODEL + m * CH + lane;
    const float* sp   = samp + (size_t)wid * 48;

    float acc = 0.0f;
#pragma unroll
    for (int j = 0; j < 16; ++j) {
        const int f = j >> 2;
        const float gx = sp[j * 3 + 0], gy = sp[j * 3 + 1], w = sp[j * 3 + 2];
        const int H = HH[f], Wd = HH[f], off = OFF[f];
        const float x = (gx + 1.0f) * (Wd * 0.5f) - 0.5f;
        const float y = (gy + 1.0f) * (H  * 0.5f) - 0.5f;
        const float x0f = floorf(x), y0f = floorf(y);
        const int   x0 = (int)x0f,   y0 = (int)y0f;
        const float wx1 = x - x0f, wy1 = y - y0f;
        const float wx0 = 1.0f - wx1, wy0 = 1.0f - wy1;
        float sv = 0.0f;
#pragma unroll
        for (int ty = 0; ty < 2; ++ty) {
            const int yi = y0 + ty;
            if (yi < 0 || yi >= H) continue;     // uniform across wave
            const float wy = ty ? wy1 : wy0;
#pragma unroll
            for (int tx = 0; tx < 2; ++tx) {
                const int xi = x0 + tx;
                if (xi < 0 || xi >= Wd) continue;
                const float wt = wy * (tx ? wx1 : wx0);
                sv = fmaf(wt, base[(size_t)(off + yi * Wd + xi) * DMODEL], sv);
            }
        }
        acc = fmaf(w, sv, acc);
    }
    // pre-projection output, row-major [L*N][D]
    pre[((size_t)l * NB + n) * DMODEL + m * CH + lane] = acc;
}

// =====================================================================
// Kernel D: output projection  out[r][o] = sum_d pre[r][d]*Wo[o][d] + bo[o]
// Same WMMA tiling as kernel A (M axis = query-row r, N axis = o).
// =====================================================================
__global__ __launch_bounds__(128) void out_proj_kernel(
    const float* __restrict__ pre, const float* __restrict__ Wo,
    const float* __restrict__ bo, float* __restrict__ out)
{
    const int r_tile = blockIdx.x;   // 38 tiles over 2400 rows
    const int o_tile = blockIdx.y;   // 4 tiles over 256 cols
    const int tid  = threadIdx.x;
    const int lane = tid & 31;
    const int wave = tid >> 5;
    const int r_base = r_tile * 64;
    const int o_base = o_tile * 64;
    const int NR = LQ * NB;          // 2400

    __shared__ float Alds[64 * 68];  // [r_local][d_local]
    __shared__ float Wlds[64 * 68];  // [o_local][d_local]

    const int row = lane & 15;
    const int kk  = (lane >> 4) << 1;

    v8f acc[4] = {v8f{}, v8f{}, v8f{}, v8f{}};

    for (int c0 = 0; c0 < DMODEL; c0 += 64) {
        __syncthreads();
        for (int idx = tid; idx < 64 * 16; idx += 128) {
            const int o_l = idx >> 4, t = idx & 15;
            const float4 w4 = *(const float4*)&Wo[(size_t)(o_base + o_l) * DMODEL + c0 + t * 4];
            *(float4*)&Wlds[o_l * 68 + t * 4] = w4;
        }
        for (int idx = tid; idx < 64 * 16; idx += 128) {
            const int r_l = idx >> 4, t = idx & 15;
            const int r = r_base + r_l;
            float4 a4 = {0.f, 0.f, 0.f, 0.f};
            if (r < NR) a4 = *(const float4*)&pre[(size_t)r * DMODEL + c0 + t * 4];
            *(float4*)&Alds[r_l * 68 + t * 4] = a4;
        }
        __syncthreads();

        const float* Abase = &Alds[(wave * 16 + row) * 68 + kk];
#pragma unroll
        for (int cc = 0; cc < 64; cc += 4) {
            const v2f a = *(const v2f*)(Abase + cc);
#pragma unroll
            for (int t = 0; t < 4; ++t) {
                const v2f b = *(const v2f*)&Wlds[(t * 16 + row) * 68 + cc + kk];
                acc[t] = __builtin_amdgcn_wmma_f32_16x16x4_f32(
                    false, a, false, b, (short)0, acc[t], false, false);
            }
        }
    }

    const int hi = (lane >> 4) * 8;
#pragma unroll
    for (int t = 0; t < 4; ++t) {
        const int o = o_base + t * 16 + row;
        const float bias = bo[o];
#pragma unroll
        for (int v = 0; v < 8; ++v) {
            const int r = r_base + wave * 16 + v + hi;
            if (r < NR) out[(size_t)r * DMODEL + o] = acc[t][v] + bias;
        }
    }
}

// =====================================================================
extern "C" void kernel_launch(void* const* d_in, const int* in_sizes, int n_in,
                              void* d_out, int out_size, void* d_ws, size_t ws_size,
                              hipStream_t stream) {
    (void)in_sizes; (void)n_in; (void)out_size; (void)ws_size;
    const float* q            = (const float*)d_in[0];
    // d_in[1] = q_pos (zeros, unused by reference)
    const float* k            = (const float*)d_in[2];
    // d_in[3] = k_pos (zeros, unused by reference)
    const unsigned char* mask = (const unsigned char*)d_in[4];
    const float* pos_centers  = (const float*)d_in[5];
    const float* valid_scales = (const float*)d_in[6];
    const float* vconv_w      = (const float*)d_in[7];
    const float* vconv_b      = (const float*)d_in[8];
    const float* out_w        = (const float*)d_in[9];
    const float* out_b        = (const float*)d_in[10];
    const float* loc_w        = (const float*)d_in[11];
    const float* loc_b        = (const float*)d_in[12];
    const float* wgt_w        = (const float*)d_in[13];
    const float* wgt_b        = (const float*)d_in[14];

    float* out  = (float*)d_out;                 // [L,N,D] = 614400 floats
    float* attn = out + (size_t)LQ * NB * DMODEL; // [N,L,FL,M,P,3] = 921600 floats

    float* ws      = (float*)d_ws;
    float* value_t = ws;                                       // N*S*D   = 27,226,112 f
    float* samp    = value_t + (size_t)NB * STOT * DMODEL;     // 19200*48 = 921,600 f
    float* pre     = samp + (size_t)LQ * NM * 48;              // 2400*256 = 614,400 f

    // 1) value conv (WMMA f32), channels-last output, mask+bias fused
    value_conv_kernel<<<dim3((STOT + 63) / 64, DMODEL / 64, NB), 128, 0, stream>>>(
        k, vconv_w, vconv_b, mask, value_t);
    // 2) sampling params: GEMV + softmax + grid transform + attn tensor
    param_kernel<<<LQ * NM, 64, 0, stream>>>(
        q, pos_centers, valid_scales, loc_w, loc_b, wgt_w, wgt_b, attn, samp);
    // 3) bilinear gather + weighted sum (wave per (l,nm), lane = channel)
    sample_kernel<<<(LQ * NM) / 8, 256, 0, stream>>>(value_t, samp, pre);
    // 4) output projection (WMMA f32)
    out_proj_kernel<<<dim3((LQ * NB + 63) / 64, DMODEL / 64), 128, 0, stream>>>(
        pre, out_w, out_b, out);
}